// MeanSquaredError3D_38474317038048
// MI455X (gfx1250) — compile-verified
//
#include <hip/hip_runtime.h>
#include <math.h>

#define NJ   24
#define COLS 14
#define HMSZ (COLS * COLS)  // 196

typedef __attribute__((ext_vector_type(2))) float v2f;
typedef __attribute__((ext_vector_type(8))) float v8f;

// limb pairs: {i00,i01,i10,i11}
__constant__ int LIMB[9][4] = {
    {0, 1, 5, 6},     {1, 2, 6, 7},     {2, 3, 7, 8},     {2, 4, 7, 9},
    {15, 16, 19, 20}, {16, 17, 20, 21}, {17, 18, 21, 22},
    {0, 23, 5, 23},   {15, 23, 19, 23}};

// global accumulator layout in d_ws:
// [0] d1 num, [1] d1 cnt, [2] d2 num, [3] d2 cnt, [4] d3 num, [5] d3 cnt,
// [6..14] S0[k], [15..23] S1[k], [24] lengV
__global__ void init_acc(float* g) {
  int t = threadIdx.x;
  if (t < 32) g[t] = 0.0f;
}

__global__ __launch_bounds__(256) void pose_loss_main(
    const float* __restrict__ o2D, const float* __restrict__ o3D,
    const float* __restrict__ h, const float* __restrict__ d,
    const float* __restrict__ t2D, const float* __restrict__ t3D,
    const float* __restrict__ v, float* __restrict__ gacc) {
  __shared__ float s_acc[25];
  __shared__ float s_x3d[NJ][3];   // unmasked 3D coords
  __shared__ float s_x3dm[NJ][3];  // depth-masked 3D coords
  __shared__ float s_vnew[NJ];
  __shared__ float s_vj[NJ];
  __shared__ float s_oob[NJ];

  const int b = blockIdx.x;
  const int tid = threadIdx.x;
  const int lane = tid & 31;
  const int wave = tid >> 5;
  const float scale = 1.0f / (float)COLS;

  if (tid < 25) s_acc[tid] = 0.0f;
  __syncthreads();

  const int hiHalf = lane >> 4;  // 0: rows 0..7, 1: rows 8..15 (WMMA D tiling)
  const int col = lane & 15;

  for (int jj = 0; jj < 3; ++jj) {
    const int j = wave + jj * 8;  // 8 waves x 3 -> joints 0..23
    const float* hp = h + ((size_t)b * NJ + j) * HMSZ;

    // ---- load heatmap in WMMA-D tiling; track argmax (first-max tiebreak)
    float hv[8];
    float best = -3.4e38f;
    int bidx = 1 << 30;
#pragma unroll
    for (int r = 0; r < 8; ++r) {
      const int row = r + 8 * hiHalf;
      const bool valid = (row < COLS) && (col < COLS);
      const int idx = row * COLS + col;
      float x = valid ? hp[idx] : 0.0f;
      hv[r] = valid ? x : 0.0f;
      if (valid) {
        if (x > best || (x == best && idx < bidx)) { best = x; bidx = idx; }
      }
    }
#pragma unroll
    for (int off = 16; off > 0; off >>= 1) {  // wave32 argmax reduce
      float ov = __shfl_xor(best, off, 32);
      int oi = __shfl_xor(bidx, off, 32);
      if (ov > best || (ov == best && oi < bidx)) { best = ov; bidx = oi; }
    }
    const int yC = bidx / COLS;
    const int xC = bidx - yC * COLS;

    // ---- per-joint targets / visibility (uniform across wave)
    const float tx = t2D[((size_t)b * NJ + j) * 2 + 0];
    const float ty = t2D[((size_t)b * NJ + j) * 2 + 1];
    const float vjf = v[((size_t)b * NJ + j) * 3];
    const bool vis = (vjf == 1.0f);
    const int mux = (int)floorf(tx * (float)COLS + 0.5f);
    const int muy = (int)floorf(ty * (float)COLS + 0.5f);
    const bool oob = vis && ((mux - 3 >= COLS) || (muy - 3 >= COLS) ||
                             (mux + 4 <= 0) || (muy + 4 <= 0));
    const float m = (vis && !oob) ? 1.0f : 0.0f;  // == place == mask1 == v_new flag

    // ---- separable clipped gaussian factors for this lane's slot
    float gyv = 0.0f, gxv = 0.0f;
    if (lane < COLS) {
      const int dy = lane - muy;
      const int dx = lane - mux;
      gyv = (dy >= -3 && dy <= 3) ? __expf(-0.5f * (float)(dy * dy)) : 0.0f;
      gxv = (dx >= -3 && dx <= 3) ? __expf(-0.5f * (float)(dx * dx)) : 0.0f;
    }

    // ---- rank-1 outer product tt = gy * gx^T via V_WMMA_F32_16X16X4_F32
    // A (16x4 f32): K=0 column = gy (lanes 0..15, VGPR0); all else zero.
    // B (4x16 f32): K=0 row    = gx (lanes 0..15, VGPR0); all else zero.
    v2f a;  a.x = (lane < 16) ? gyv : 0.0f;  a.y = 0.0f;
    v2f bm; bm.x = (lane < 16) ? gxv : 0.0f; bm.y = 0.0f;
    v8f c = {0.0f, 0.0f, 0.0f, 0.0f, 0.0f, 0.0f, 0.0f, 0.0f};
    v8f tt = __builtin_amdgcn_wmma_f32_16x16x4_f32(
        false, a, false, bm, (short)0, c, false, false);

    // ---- masked heatmap SSE; tt layout matches hv layout exactly
    float p = 0.0f;
#pragma unroll
    for (int r = 0; r < 8; ++r) {
      const float diff = hv[r] - tt[r];
      p += diff * diff;
    }
#pragma unroll
    for (int off = 16; off > 0; off >>= 1) p += __shfl_xor(p, off, 32);

    if (lane == 0) {
      atomicAdd(&s_acc[0], m * p);  // d1 numerator (mask zeroes both h and tt terms)
      atomicAdd(&s_acc[1], m);      // d1 count (visible, in-bounds joints)

      // gather offsets at argmax, build coords
      const float xCs = (float)xC * scale, yCs = (float)yC * scale;
      const size_t o2i = ((size_t)b * (2 * NJ) + j) * HMSZ + yC * COLS + xC;
      const float x2d0 = o2D[o2i] + xCs;
      const float x2d1 = o2D[o2i + (size_t)NJ * HMSZ] + yCs;
      const size_t o3i = ((size_t)b * (3 * NJ) + j) * HMSZ + yC * COLS + xC;
      const float x30 = o3D[o3i] + xCs;
      const float x31 = o3D[o3i + (size_t)NJ * HMSZ] + yCs;
      const float x32 = o3D[o3i + (size_t)(2 * NJ) * HMSZ];

      const float e0 = (x2d0 - tx) * m;
      const float e1 = (x2d1 - ty) * m;
      atomicAdd(&s_acc[2], e0 * e0 + e1 * e1);  // d2 numerator
      atomicAdd(&s_acc[3], m);                  // N2 partial (sum(v_new)/3)

      s_x3d[j][0] = x30; s_x3d[j][1] = x31; s_x3d[j][2] = x32;
      s_vnew[j] = m;
      s_vj[j] = vjf;
      s_oob[j] = oob ? 1.0f : 0.0f;
    }
  }
  __syncthreads();

  // ---- cross-joint terms
  bool anyo = false;
#pragma unroll
  for (int jq = 0; jq < NJ; ++jq) anyo |= (s_oob[jq] != 0.0f);
  const bool dok = d[b] > -990.0f;
  const float df = dok ? 1.0f : 0.0f;

  if (tid < NJ) {
    const float x0 = s_x3d[tid][0] * df;  // x3D = where(d_ok, x3D_full, 0)
    const float x1 = s_x3d[tid][1] * df;
    const float x2 = s_x3d[tid][2] * df;
    s_x3dm[tid][0] = x0; s_x3dm[tid][1] = x1; s_x3dm[tid][2] = x2;
    const float v3f = (dok && !anyo) ? s_vj[tid] : 0.0f;  // v3D uses ORIGINAL v
    const float* t3 = t3D + ((size_t)b * NJ + tid) * 3;
    const float e0 = (x0 - t3[0]) * v3f;
    const float e1 = (x1 - t3[1]) * v3f;
    const float e2 = (x2 - t3[2]) * v3f;
    atomicAdd(&s_acc[4], e0 * e0 + e1 * e1 + e2 * e2);  // d3 numerator
    atomicAdd(&s_acc[5], v3f);                          // N3 partial
  }
  __syncthreads();

  if (tid < 9) {  // limb symmetry partial sums (vv in {0,1} so vv^2 == vv)
    const int i00 = LIMB[tid][0], i01 = LIMB[tid][1];
    const int i10 = LIMB[tid][2], i11 = LIMB[tid][3];
    const float vv = s_vnew[i00] * s_vnew[i01] * s_vnew[i10] * s_vnew[i11];
    float l0 = 0.0f, l1 = 0.0f;
#pragma unroll
    for (int cix = 0; cix < 3; ++cix) {
      const float u0 = s_x3dm[i00][cix] - s_x3dm[i01][cix];
      const float u1 = s_x3dm[i10][cix] - s_x3dm[i11][cix];
      l0 += u0 * u0;
      l1 += u1 * u1;
    }
    atomicAdd(&s_acc[6 + tid], vv * l0);   // S0[k]
    atomicAdd(&s_acc[15 + tid], vv * l1);  // S1[k]
    atomicAdd(&s_acc[24], vv);             // lengV partial
  }
  __syncthreads();

  if (tid < 25) atomicAdd(&gacc[tid], s_acc[tid]);
}

__global__ void finalize_loss(const float* __restrict__ g, float* __restrict__ out) {
  if (threadIdx.x == 0 && blockIdx.x == 0) {
    const float d1 = g[0] / g[1];
    const float d2 = g[2] / g[3];
    const float d3 = g[4] / g[5];
    float ll = 0.0f;
#pragma unroll
    for (int k = 0; k < 9; ++k) {
      const float s0 = sqrtf(g[6 + k]);
      const float s1 = sqrtf(g[15 + k]);
      const float dl = s0 - s1;
      ll += dl * dl;
    }
    out[0] = d1 + d2 + d3 + ll / g[24];
  }
}

extern "C" void kernel_launch(void* const* d_in, const int* in_sizes, int n_in,
                              void* d_out, int out_size, void* d_ws, size_t ws_size,
                              hipStream_t stream) {
  (void)n_in; (void)out_size; (void)ws_size;
  const float* o2D = (const float*)d_in[0];
  const float* o3D = (const float*)d_in[1];
  const float* h   = (const float*)d_in[2];
  const float* dpt = (const float*)d_in[3];
  const float* t2D = (const float*)d_in[4];
  const float* t3D = (const float*)d_in[5];
  const float* v   = (const float*)d_in[6];
  float* gacc = (float*)d_ws;
  float* out  = (float*)d_out;
  const int B = in_sizes[3];  // d is [B]

  init_acc<<<1, 32, 0, stream>>>(gacc);
  pose_loss_main<<<B, 256, 0, stream>>>(o2D, o3D, h, dpt, t2D, t3D, v, gacc);
  finalize_loss<<<1, 1, 0, stream>>>(gacc, out);
}